// Sinkhorn_4836133175362
// MI455X (gfx1250) — compile-verified
//
#include <hip/hip_runtime.h>
#include <math.h>

typedef __attribute__((ext_vector_type(2))) float v2f;
typedef __attribute__((ext_vector_type(8))) float v8f;

#define N 4096
#define D 64
#define EPS 3.0f
#define INV_EPS (1.0f / 3.0f)
// eps * log(n) = 3 * log(4096)
#define EPS_LOGN 24.95329850015803f
#define NEG_INF (-__builtin_huge_valf())
#define NCHUNK 32            // row-chunks for g-update stage 1
#define CHUNK_ROWS (N / NCHUNK)  // 128

// ---------------------------------------------------------------------------
// 0.5*|x_i|^2 and 0.5*|y_j|^2
// ---------------------------------------------------------------------------
__global__ void norms_kernel(const float* __restrict__ X,
                             const float* __restrict__ Y,
                             float* __restrict__ xn, float* __restrict__ yn) {
    int i = blockIdx.x * blockDim.x + threadIdx.x;
    if (i >= N) return;
    float sx = 0.f, sy = 0.f;
    const float* xr = X + (size_t)i * D;
    const float* yr = Y + (size_t)i * D;
#pragma unroll
    for (int k = 0; k < D; ++k) {
        sx += xr[k] * xr[k];
        sy += yr[k] * yr[k];
    }
    xn[i] = 0.5f * sx;
    yn[i] = 0.5f * sy;
}

// ---------------------------------------------------------------------------
// S[i][j] = -C_ij/eps = (x_i . y_j - 0.5|x_i|^2 - 0.5|y_j|^2) / eps
// One wave32 per 16x16 tile; fp32 WMMA 16x16x4, 16 steps over K=64.
// ---------------------------------------------------------------------------
__global__ void gram_wmma_kernel(const float* __restrict__ X,
                                 const float* __restrict__ Y,
                                 const float* __restrict__ xn,
                                 const float* __restrict__ yn,
                                 float* __restrict__ S) {
    const int lane = threadIdx.x & 31;
    const int wv   = threadIdx.x >> 5;
    const int tile = blockIdx.x * 8 + wv;        // 65536 tiles, 8 waves/block
    const int ti = tile >> 8;                    // 256 tile-cols
    const int tj = tile & 255;

    const int mrow = lane & 15;                  // A row (M) / B col (N)
    const int kb   = (lane >> 4) << 1;           // K sub-offset: 0 or 2

    const float* xrow = X + (size_t)(ti * 16 + mrow) * D;
    const float* yrow = Y + (size_t)(tj * 16 + mrow) * D;

    v8f acc = {};
#pragma unroll
    for (int k0 = 0; k0 < D; k0 += 4) {
        v2f a, b;
        a.x = xrow[k0 + kb];
        a.y = xrow[k0 + kb + 1];
        b.x = yrow[k0 + kb];
        b.y = yrow[k0 + kb + 1];
        // D = A(16x4) * B(4x16) + C, full fp32
        acc = __builtin_amdgcn_wmma_f32_16x16x4_f32(
            false, a, false, b, (short)0, acc, false, false);
    }

    // D layout: VGPR v -> row = v + 8*(lane/16), col = lane&15
    const int col = lane & 15;
    const int j   = tj * 16 + col;
    const float ynj = yn[j];
    const int rbase = ti * 16 + ((lane >> 4) << 3);
#pragma unroll
    for (int v = 0; v < 8; ++v) {
        const int i = rbase + v;
        S[(size_t)i * N + j] = (acc[v] - xn[i] - ynj) * INV_EPS;
    }
}

// ---------------------------------------------------------------------------
__global__ void init_fg_kernel(float* __restrict__ f, float* __restrict__ g) {
    int i = blockIdx.x * blockDim.x + threadIdx.x;
    if (i < N) { f[i] = 0.f; g[i] = 0.f; }
}

__device__ __forceinline__ void lse_push(float z, float& m, float& s) {
    if (z <= m) {
        s += __expf(z - m);
    } else {
        s = s * __expf(m - z) + 1.0f;
        m = z;
    }
}

// ---------------------------------------------------------------------------
// f_i = eps*log(n) - eps * logsumexp_j( g_j/eps + S_ij ).  One block per row.
// ---------------------------------------------------------------------------
__global__ void f_update_kernel(const float* __restrict__ S,
                                const float* __restrict__ g,
                                float* __restrict__ f) {
    const int i = blockIdx.x;
    const int t = threadIdx.x;                   // 256 threads
    const float* row = S + (size_t)i * N;

    float m = NEG_INF, s = 0.f;
#pragma unroll 4
    for (int j = t; j < N; j += 256) {
        lse_push(__builtin_fmaf(g[j], INV_EPS, row[j]), m, s);
    }

    __shared__ float sm[256];
    __shared__ float ss[256];
    sm[t] = m; ss[t] = s;
    __syncthreads();
    for (int off = 128; off > 0; off >>= 1) {
        if (t < off) {
            float m2 = sm[t + off], s2 = ss[t + off];
            float m1 = sm[t],       s1 = ss[t];
            float mm = fmaxf(m1, m2);
            ss[t] = s1 * __expf(m1 - mm) + s2 * __expf(m2 - mm);
            sm[t] = mm;
        }
        __syncthreads();
    }
    if (t == 0) f[i] = EPS_LOGN - EPS * (sm[0] + __logf(ss[0]));
}

// ---------------------------------------------------------------------------
// g-update stage 1: partial logsumexp down a 128-row chunk of each column.
// grid (N/256, NCHUNK); coalesced column streaming of row-major S.
// ---------------------------------------------------------------------------
__global__ void g_partial_kernel(const float* __restrict__ S,
                                 const float* __restrict__ f,
                                 float* __restrict__ pm,
                                 float* __restrict__ ps) {
    const int j     = blockIdx.x * 256 + threadIdx.x;
    const int chunk = blockIdx.y;
    const int i0    = chunk * CHUNK_ROWS;

    __shared__ float fsh[CHUNK_ROWS];
    if (threadIdx.x < CHUNK_ROWS) fsh[threadIdx.x] = f[i0 + threadIdx.x] * INV_EPS;
    __syncthreads();

    float m = NEG_INF, s = 0.f;
    const float* col = S + (size_t)i0 * N + j;
#pragma unroll 4
    for (int r = 0; r < CHUNK_ROWS; ++r) {
        lse_push(fsh[r] + col[(size_t)r * N], m, s);
    }
    pm[chunk * N + j] = m;
    ps[chunk * N + j] = s;
}

// g-update stage 2: merge NCHUNK partials per column.
__global__ void g_merge_kernel(const float* __restrict__ pm,
                               const float* __restrict__ ps,
                               float* __restrict__ g) {
    const int j = blockIdx.x * blockDim.x + threadIdx.x;
    if (j >= N) return;
    float m = NEG_INF;
#pragma unroll
    for (int c = 0; c < NCHUNK; ++c) m = fmaxf(m, pm[c * N + j]);
    float s = 0.f;
#pragma unroll
    for (int c = 0; c < NCHUNK; ++c) s += ps[c * N + j] * __expf(pm[c * N + j] - m);
    g[j] = EPS_LOGN - EPS * (m + __logf(s));
}

// ---------------------------------------------------------------------------
__global__ void finalize_kernel(const float* __restrict__ f,
                                const float* __restrict__ g,
                                float* __restrict__ out) {
    int i = blockIdx.x * blockDim.x + threadIdx.x;
    if (i < N) out[i] = f[i];
    else if (i < 2 * N) out[i] = g[i - N];
}

// ---------------------------------------------------------------------------
extern "C" void kernel_launch(void* const* d_in, const int* in_sizes, int n_in,
                              void* d_out, int out_size, void* d_ws, size_t ws_size,
                              hipStream_t stream) {
    const float* X = (const float*)d_in[0];
    const float* Y = (const float*)d_in[1];
    float* out = (float*)d_out;

    char* ws = (char*)d_ws;
    float* S  = (float*)ws;                               // 4096*4096 f32 = 64 MB
    float* xn = (float*)(ws + (size_t)N * N * sizeof(float));
    float* yn = xn + N;
    float* f  = yn + N;
    float* g  = f + N;
    float* pm = g + N;                                    // NCHUNK * N
    float* ps = pm + NCHUNK * N;                          // NCHUNK * N

    norms_kernel<<<N / 256, 256, 0, stream>>>(X, Y, xn, yn);
    // 256x256 tiles of 16x16, 8 waves (tiles) per 256-thread block
    gram_wmma_kernel<<<(256 * 256) / 8, 256, 0, stream>>>(X, Y, xn, yn, S);
    init_fg_kernel<<<N / 256, 256, 0, stream>>>(f, g);

    for (int it = 0; it < 50; ++it) {
        f_update_kernel<<<N, 256, 0, stream>>>(S, g, f);
        g_partial_kernel<<<dim3(N / 256, NCHUNK), 256, 0, stream>>>(S, f, pm, ps);
        g_merge_kernel<<<N / 256, 256, 0, stream>>>(pm, ps, g);
    }

    finalize_kernel<<<(2 * N) / 256, 256, 0, stream>>>(f, g, out);
}